// HeteroGTNLite_30365418783291
// MI455X (gfx1250) — compile-verified
//
#include <hip/hip_runtime.h>
#include <stdint.h>

typedef __attribute__((ext_vector_type(2))) float v2f;
typedef __attribute__((ext_vector_type(8))) float v8f;
typedef int vint4 __attribute__((vector_size(16)));   // matches builtin param pointee

#define NTX   100000
#define NADDR 100000
#define NE    1600000
#define D     128
#define ROWP  132                 // padded LDS row stride (floats): conflict-free A reads

#if defined(__gfx1250__) && __has_builtin(__builtin_amdgcn_global_load_async_to_lds_b128)
#define HAVE_ASYNC_LDS 1
#else
#define HAVE_ASYNC_LDS 0
#endif

// ---------------- workspace layout (float offsets) ----------------
constexpr size_t OFF_RW     = 0;                       // 8
constexpr size_t OFF_B0TX   = 8;                       // 128
constexpr size_t OFF_B0AD   = 136;                     // 128
constexpr size_t OFF_B1TX   = 264;                     // 16
constexpr size_t OFF_B1AD   = 280;                     // 16
constexpr size_t OFF_WLS0   = 512;                     // 4*128*128
constexpr size_t OFF_WRC0TX = OFF_WLS0   + 65536;      // 128*128
constexpr size_t OFF_WRC0AD = OFF_WRC0TX + 16384;      // 128*128
constexpr size_t OFF_WLS1   = OFF_WRC0AD + 16384;      // 4*128*16 (N padded to 16)
constexpr size_t OFF_WRC1TX = OFF_WLS1   + 8192;       // 128*16
constexpr size_t OFF_WRC1AD = OFF_WRC1TX + 2048;       // 128*16
constexpr size_t OFF_CNT    = 111104;                  // 4*100000 (becomes 1/cnt)
constexpr size_t OFF_AGGA   = OFF_CNT  + 400000;
constexpr size_t OFF_AGGB   = OFF_AGGA + (size_t)NTX * D;
constexpr size_t OFF_HTX    = OFF_AGGB + (size_t)NTX * D;
constexpr size_t OFF_HAD    = OFF_HTX  + (size_t)NTX * D;

// ---------------- tiny kernels ----------------
__global__ void k_rw(const float* __restrict__ rel0, const float* __restrict__ rel1,
                     float* __restrict__ rw) {
  if (threadIdx.x == 0 && blockIdx.x == 0) {
    for (int l = 0; l < 2; ++l) {
      const float* r = l ? rel1 : rel0;
      float m = r[0];
      for (int i = 1; i < 4; ++i) m = fmaxf(m, r[i]);
      float e[4], s = 0.f;
      for (int i = 0; i < 4; ++i) { e[i] = expf(r[i] - m); s += e[i]; }
      for (int i = 0; i < 4; ++i) rw[l * 4 + i] = e[i] / s;
    }
  }
}

// Build rel-weighted / combined weight matrices and biases.
__global__ void k_prescale(const float* __restrict__ Wl0, const float* __restrict__ bl0,
                           const float* __restrict__ Wr0, const float* __restrict__ Wl1,
                           const float* __restrict__ bl1, const float* __restrict__ Wr1,
                           float* __restrict__ ws) {
  const float* rw = ws + OFF_RW;
  int i = blockIdx.x * blockDim.x + threadIdx.x;
  if (i < 65536) {                               // WlS0[t][k][n] = rw0[t]*Wl0
    int t = i >> 14;
    ws[OFF_WLS0 + i] = rw[t] * Wl0[i];
    return;
  }
  i -= 65536;
  if (i < 16384) { ws[OFF_WRC0TX + i] = rw[0] * Wr0[0 * 16384 + i] + rw[2] * Wr0[2 * 16384 + i]; return; }
  i -= 16384;
  if (i < 16384) { ws[OFF_WRC0AD + i] = rw[1] * Wr0[1 * 16384 + i] + rw[3] * Wr0[3 * 16384 + i]; return; }
  i -= 16384;
  if (i < 8192) {                                // WlS1 padded N=2 -> 16
    int t = i >> 11, r = i & 2047, k = r >> 4, n = r & 15;
    ws[OFF_WLS1 + i] = (n < 2) ? rw[4 + t] * Wl1[t * 256 + k * 2 + n] : 0.0f;
    return;
  }
  i -= 8192;
  if (i < 2048) { int k = i >> 4, n = i & 15;
    ws[OFF_WRC1TX + i] = (n < 2) ? (rw[4] * Wr1[0 * 256 + k * 2 + n] + rw[6] * Wr1[2 * 256 + k * 2 + n]) : 0.0f;
    return; }
  i -= 2048;
  if (i < 2048) { int k = i >> 4, n = i & 15;
    ws[OFF_WRC1AD + i] = (n < 2) ? (rw[5] * Wr1[1 * 256 + k * 2 + n] + rw[7] * Wr1[3 * 256 + k * 2 + n]) : 0.0f;
    return; }
  i -= 2048;
  if (i < 128) { ws[OFF_B0TX + i] = rw[0] * bl0[0 * 128 + i] + rw[2] * bl0[2 * 128 + i]; return; }
  i -= 128;
  if (i < 128) { ws[OFF_B0AD + i] = rw[1] * bl0[1 * 128 + i] + rw[3] * bl0[3 * 128 + i]; return; }
  i -= 128;
  if (i < 16) { ws[OFF_B1TX + i] = (i < 2) ? (rw[4] * bl1[0 * 2 + i] + rw[6] * bl1[2 * 2 + i]) : 0.0f; return; }
  i -= 16;
  if (i < 16) { ws[OFF_B1AD + i] = (i < 2) ? (rw[5] * bl1[1 * 2 + i] + rw[7] * bl1[3 * 2 + i]) : 0.0f; return; }
}

// ---------------- degree counts ----------------
__global__ void k_count(const int* __restrict__ dtt, const int* __restrict__ daa,
                        const int* __restrict__ dat, const int* __restrict__ dta,
                        float* __restrict__ cnt) {
  int i = blockIdx.x * blockDim.x + threadIdx.x;
  if (i >= NE) return;
  int t = blockIdx.y;
  const int* d = (t == 0) ? dtt : (t == 1) ? daa : (t == 2) ? dat : dta;
  atomicAdd(&cnt[(size_t)t * 100000 + d[i]], 1.0f);
}

__global__ void k_invcnt(float* __restrict__ cnt) {
  int i = blockIdx.x * blockDim.x + threadIdx.x;
  if (i < 400000) cnt[i] = 1.0f / fmaxf(cnt[i], 1.0f);
}

// ---------------- edge scatter: agg[dst] += x[src] ----------------
__global__ void k_scatter(const int* __restrict__ esrc, const int* __restrict__ edst,
                          const float* __restrict__ xsrc, float* __restrict__ agg) {
  int tid = blockIdx.x * blockDim.x + threadIdx.x;     // E*32 threads, 4 floats each
  if (tid >= NE * 32) return;
  int e  = tid >> 5;
  int f4 = (tid & 31) * 4;
  int s = esrc[e], d = edst[e];
  float4 v = *(const float4*)(xsrc + (size_t)s * D + f4);
  float* a = agg + (size_t)d * D + f4;
  atomicAdd(a + 0, v.x);
  atomicAdd(a + 1, v.y);
  atomicAdd(a + 2, v.z);
  atomicAdd(a + 3, v.w);
}

// ---------------- WMMA helpers ----------------
// A fragments from LDS (pre-scaled rows), B fragments from global weights.
__device__ __forceinline__ void mma_lds(const float* __restrict__ aRow,      // &smem[row base]
                                        const float* __restrict__ W, int ldw,
                                        int koff, int ncol, v8f& c) {
#pragma unroll
  for (int k0 = 0; k0 < D; k0 += 4) {
    int k = k0 + koff;                    // lanes 0-15: K=k0,k0+1; lanes 16-31: K=k0+2,k0+3
    v2f a, b;
    a.x = aRow[k];                        // ds_load_b64
    a.y = aRow[k + 1];
    b.x = W[(size_t)k * ldw + ncol];
    b.y = W[(size_t)(k + 1) * ldw + ncol];
    c = __builtin_amdgcn_wmma_f32_16x16x4_f32(false, a, false, b, (short)0, c, false, false);
  }
}

// A fragments straight from global (layer-1, single wave per block).
__device__ __forceinline__ void mma_src(const float* __restrict__ A, const float* __restrict__ inv,
                                        const float* __restrict__ W, int ldw,
                                        int row, int koff, int ncol, v8f& c) {
  float scale = inv ? inv[row] : 1.0f;
  const float* ap = A + (size_t)row * D;
  __builtin_prefetch(ap, 0, 3);           // global_prefetch_b8
#pragma unroll
  for (int k0 = 0; k0 < D; k0 += 4) {
    int k = k0 + koff;
    v2f a, b;
    a.x = ap[k] * scale;
    a.y = ap[k + 1] * scale;
    b.x = W[(size_t)k * ldw + ncol];
    b.y = W[(size_t)(k + 1) * ldw + ncol];
    c = __builtin_amdgcn_wmma_f32_16x16x4_f32(false, a, false, b, (short)0, c, false, false);
  }
}

// ---- layer-0 fused GEMM: LDS-staged A tiles shared by all 8 waves of the block ----
__global__ void __launch_bounds__(256) k_gemm_l0(
    const float* __restrict__ aggA, const float* __restrict__ invA,
    const float* __restrict__ aggB, const float* __restrict__ invB,
    const float* __restrict__ xdst,
    const float* __restrict__ WlA, const float* __restrict__ WlB,
    const float* __restrict__ WrC, const float* __restrict__ bias,
    float* __restrict__ out) {
  __shared__ __align__(16) float smem[3 * 16 * ROWP];  // aggA | aggB | xdst tiles

  int lane = threadIdx.x & 31;
  int wave = threadIdx.x >> 5;
  int m0 = blockIdx.x * 16;
  int n0 = wave * 16;
  int koff = (lane >> 4) * 2;
  int ncol = n0 + (lane & 15);

  // Stage aggA (s=0) and aggB (s=1), folding in the 1/cnt mean-division.
#pragma unroll
  for (int q = 0; q < 4; ++q) {
    int idx = threadIdx.x + q * 256;                   // 0..1023
    int s = idx >> 9;
    int r = (idx >> 5) & 15;
    int c4 = (idx & 31) * 4;
    const float* src = s ? aggB : aggA;
    const float* inv = s ? invB : invA;
    float sc = inv[m0 + r];
    float4 v = *(const float4*)(src + (size_t)(m0 + r) * D + c4);
    float4 sv = make_float4(v.x * sc, v.y * sc, v.z * sc, v.w * sc);
    *(float4*)&smem[s * 16 * ROWP + r * ROWP + c4] = sv;
  }

  // Stage xdst (s=2): async global->LDS DMA when available.
#if HAVE_ASYNC_LDS
  {
    int idx = threadIdx.x;
#pragma unroll
    for (int q = 0; q < 2; ++q, idx += 256) {          // 0..511 -> 16 rows x 32 float4
      int r = idx >> 5;
      int c4 = (idx & 31) * 4;
      const float* g = xdst + (size_t)(m0 + r) * D + c4;
      float* l = &smem[2 * 16 * ROWP + r * ROWP + c4];
      // AS(3) pointer value = low 32 bits of the generic shared address (LDS offset).
      __builtin_amdgcn_global_load_async_to_lds_b128(
          (__attribute__((address_space(1))) vint4*)(uintptr_t)g,
          (__attribute__((address_space(3))) vint4*)(uint32_t)(uintptr_t)l,
          0, 0);
    }
    asm volatile("s_wait_asynccnt 0x0" ::: "memory");
  }
#else
  {
    int idx = threadIdx.x;
#pragma unroll
    for (int q = 0; q < 2; ++q, idx += 256) {
      int r = idx >> 5;
      int c4 = (idx & 31) * 4;
      float4 v = *(const float4*)(xdst + (size_t)(m0 + r) * D + c4);
      *(float4*)&smem[2 * 16 * ROWP + r * ROWP + c4] = v;
    }
  }
#endif
  __syncthreads();

  v8f c;
  float bv = bias[ncol];
#pragma unroll
  for (int r = 0; r < 8; ++r) c[r] = bv;

  const float* aRow = &smem[(lane & 15) * ROWP];
  mma_lds(aRow + 0 * 16 * ROWP, WlA, D, koff, ncol, c);
  mma_lds(aRow + 1 * 16 * ROWP, WlB, D, koff, ncol, c);
  mma_lds(aRow + 2 * 16 * ROWP, WrC, D, koff, ncol, c);

  int rbase = m0 + (lane >> 4) * 8;                    // C layout: VGPR r -> M = r + 8*(lane>=16)
#pragma unroll
  for (int r = 0; r < 8; ++r)
    out[(size_t)(rbase + r) * D + ncol] = fmaxf(c[r], 0.0f);   // ReLU
}

// ---- layer-1 fused GEMM (OUT=2 padded to N=16, one wave per row tile) ----
__global__ void __launch_bounds__(32) k_gemm_l1(
    const float* __restrict__ aggA, const float* __restrict__ invA,
    const float* __restrict__ aggB, const float* __restrict__ invB,
    const float* __restrict__ xdst,
    const float* __restrict__ WlA, const float* __restrict__ WlB,
    const float* __restrict__ WrC, const float* __restrict__ bias,
    float* __restrict__ out) {
  int lane = threadIdx.x & 31;
  int m0 = blockIdx.x * 16;
  int row  = m0 + (lane & 15);
  int koff = (lane >> 4) * 2;
  int ncol = lane & 15;
  v8f c;
  float bv = bias[ncol];
#pragma unroll
  for (int r = 0; r < 8; ++r) c[r] = bv;
  mma_src(aggA, invA,    WlA, 16, row, koff, ncol, c);
  mma_src(aggB, invB,    WlB, 16, row, koff, ncol, c);
  mma_src(xdst, nullptr, WrC, 16, row, koff, ncol, c);
  if (ncol < 2) {
    int rbase = m0 + (lane >> 4) * 8;
#pragma unroll
    for (int r = 0; r < 8; ++r)
      out[(size_t)(rbase + r) * 2 + ncol] = c[r];
  }
}

// ---------------- launcher ----------------
extern "C" void kernel_launch(void* const* d_in, const int* in_sizes, int n_in,
                              void* d_out, int out_size, void* d_ws, size_t ws_size,
                              hipStream_t stream) {
  (void)in_sizes; (void)n_in; (void)out_size; (void)ws_size;
  const float* x_tx   = (const float*)d_in[0];
  const float* x_addr = (const float*)d_in[1];
  const float* Wl0  = (const float*)d_in[2];
  const float* bl0  = (const float*)d_in[3];
  const float* Wr0  = (const float*)d_in[4];
  const float* rel0 = (const float*)d_in[5];
  const float* Wl1  = (const float*)d_in[6];
  const float* bl1  = (const float*)d_in[7];
  const float* Wr1  = (const float*)d_in[8];
  const float* rel1 = (const float*)d_in[9];
  const int* e_tt = (const int*)d_in[10];
  const int* e_aa = (const int*)d_in[11];
  const int* e_at = (const int*)d_in[12];
  const int* e_ta = (const int*)d_in[13];
  float* ws  = (float*)d_ws;
  float* out = (float*)d_out;

  float* cnt  = ws + OFF_CNT;
  float* aggA = ws + OFF_AGGA;
  float* aggB = ws + OFF_AGGB;
  float* h_tx = ws + OFF_HTX;
  float* h_ad = ws + OFF_HAD;
  const size_t aggBytes = (size_t)NTX * D * sizeof(float);

  // 1) rel softmax + weight prescale
  k_rw<<<1, 32, 0, stream>>>(rel0, rel1, ws + OFF_RW);
  k_prescale<<<(110880 + 255) / 256, 256, 0, stream>>>(Wl0, bl0, Wr0, Wl1, bl1, Wr1, ws);

  // 2) per-type in-degree -> reciprocal
  (void)hipMemsetAsync(cnt, 0, 400000 * sizeof(float), stream);
  dim3 gcnt((NE + 255) / 256, 4);
  k_count<<<gcnt, 256, 0, stream>>>(e_tt + NE, e_aa + NE, e_at + NE, e_ta + NE, cnt);
  k_invcnt<<<(400000 + 255) / 256, 256, 0, stream>>>(cnt);

  dim3 gsc((NE * 32 + 255) / 256);

  // 3) layer 0, tx dst: edge types tt(0) and at(2)
  (void)hipMemsetAsync(aggA, 0, aggBytes, stream);
  (void)hipMemsetAsync(aggB, 0, aggBytes, stream);
  k_scatter<<<gsc, 256, 0, stream>>>(e_tt, e_tt + NE, x_tx,   aggA);
  k_scatter<<<gsc, 256, 0, stream>>>(e_at, e_at + NE, x_addr, aggB);
  k_gemm_l0<<<NTX / 16, 256, 0, stream>>>(aggA, cnt + 0, aggB, cnt + 200000, x_tx,
      ws + OFF_WLS0 + 0 * 16384, ws + OFF_WLS0 + 2 * 16384,
      ws + OFF_WRC0TX, ws + OFF_B0TX, h_tx);

  // 4) layer 0, addr dst: edge types aa(1) and ta(3)
  (void)hipMemsetAsync(aggA, 0, aggBytes, stream);
  (void)hipMemsetAsync(aggB, 0, aggBytes, stream);
  k_scatter<<<gsc, 256, 0, stream>>>(e_aa, e_aa + NE, x_addr, aggA);
  k_scatter<<<gsc, 256, 0, stream>>>(e_ta, e_ta + NE, x_tx,   aggB);
  k_gemm_l0<<<NADDR / 16, 256, 0, stream>>>(aggA, cnt + 100000, aggB, cnt + 300000, x_addr,
      ws + OFF_WLS0 + 1 * 16384, ws + OFF_WLS0 + 3 * 16384,
      ws + OFF_WRC0AD, ws + OFF_B0AD, h_ad);

  // 5) layer 1, tx dst -> d_out[0 : 200000]
  (void)hipMemsetAsync(aggA, 0, aggBytes, stream);
  (void)hipMemsetAsync(aggB, 0, aggBytes, stream);
  k_scatter<<<gsc, 256, 0, stream>>>(e_tt, e_tt + NE, h_tx, aggA);
  k_scatter<<<gsc, 256, 0, stream>>>(e_at, e_at + NE, h_ad, aggB);
  k_gemm_l1<<<NTX / 16, 32, 0, stream>>>(aggA, cnt + 0, aggB, cnt + 200000, h_tx,
      ws + OFF_WLS1 + 0 * 2048, ws + OFF_WLS1 + 2 * 2048,
      ws + OFF_WRC1TX, ws + OFF_B1TX, out);

  // 6) layer 1, addr dst -> d_out[200000 : 400000]
  (void)hipMemsetAsync(aggA, 0, aggBytes, stream);
  (void)hipMemsetAsync(aggB, 0, aggBytes, stream);
  k_scatter<<<gsc, 256, 0, stream>>>(e_aa, e_aa + NE, h_ad, aggA);
  k_scatter<<<gsc, 256, 0, stream>>>(e_ta, e_ta + NE, h_tx, aggB);
  k_gemm_l1<<<NADDR / 16, 32, 0, stream>>>(aggA, cnt + 100000, aggB, cnt + 300000, h_ad,
      ws + OFF_WLS1 + 1 * 2048, ws + OFF_WLS1 + 3 * 2048,
      ws + OFF_WRC1AD, ws + OFF_B1AD, out + 200000);
}